// RKDLoss_81003083202777
// MI455X (gfx1250) — compile-verified
//
#include <hip/hip_runtime.h>

typedef __attribute__((ext_vector_type(2))) float v2f;
typedef __attribute__((ext_vector_type(8))) float v8f;

#define N_EMB 1024
#define D_EMB 512
#define NORM_EPS 1e-12f

// ---------------------------------------------------------------------------
// Kernel 1: per-row squared norms + inverse norms for both embeddings.
// One wave32 per row (2048 rows total: 1024 student + 1024 teacher).
// Also zeroes the global loss accumulator (stream order guarantees it lands
// before kernel 2's atomics).
// ---------------------------------------------------------------------------
__global__ void rkd_rowstats(const float* __restrict__ s,
                             const float* __restrict__ t,
                             float* __restrict__ sq_s, float* __restrict__ sq_t,
                             float* __restrict__ in_s, float* __restrict__ in_t,
                             float* __restrict__ acc) {
    if (blockIdx.x == 0 && threadIdx.x == 0) acc[0] = 0.0f;

    const int wid  = blockIdx.x * 8 + (threadIdx.x >> 5);   // 8 waves / block
    const int lane = threadIdx.x & 31;

    const float* src;
    float *sqo, *ino;
    int row;
    if (wid < N_EMB) { src = s; row = wid;         sqo = sq_s; ino = in_s; }
    else             { src = t; row = wid - N_EMB; sqo = sq_t; ino = in_t; }

    float sum = 0.0f;
    for (int k = lane; k < D_EMB; k += 32) {
        float v = src[row * D_EMB + k];
        sum = fmaf(v, v, sum);
    }
#pragma unroll
    for (int off = 16; off > 0; off >>= 1)
        sum += __shfl_xor(sum, off, 32);

    if (lane == 0) {
        sqo[row] = sum;
        ino[row] = 1.0f / fmaxf(sqrtf(sum), NORM_EPS);   // F.normalize eps
    }
}

// ---------------------------------------------------------------------------
// Kernel 2: one wave32 per 16x16 Gram tile (upper triangle of the 64x64 tile
// grid; off-diagonal tiles contribute twice by symmetry). Two f32 WMMA
// accumulators (student + teacher Gram tiles), K-loop in steps of 4 using
// V_WMMA_F32_16X16X4_F32. Loss epilogue consumed in-register, wave-reduced,
// one atomicAdd per tile.
//
// A-fragment layout (ISA 16x4 f32): lanes 0-15 -> M=lane, K={0,1} in v[0..1];
// lanes 16-31 -> M=lane-16, K={2,3}. For a Gram (B = E^T), the B fragment has
// the identical addressing with the column tile base: one float2 load each.
// ---------------------------------------------------------------------------
__global__ void rkd_gram_tile_loss(const float* __restrict__ S,
                                   const float* __restrict__ T,
                                   const float* __restrict__ sq_s,
                                   const float* __restrict__ sq_t,
                                   const float* __restrict__ in_s,
                                   const float* __restrict__ in_t,
                                   float* __restrict__ acc) {
    const int bi = blockIdx.y;           // row tile
    const int bj = blockIdx.x;           // col tile
    if (bj < bi) return;                 // uniform: whole wave exits together

    const int lane = threadIdx.x;        // 0..31 (one wave per block)
    const int m    = lane & 15;
    const int kk   = (lane >> 4) << 1;   // 0 or 2

    const float* aS = S + (bi * 16 + m) * D_EMB + kk;
    const float* bS = S + (bj * 16 + m) * D_EMB + kk;
    const float* aT = T + (bi * 16 + m) * D_EMB + kk;
    const float* bT = T + (bj * 16 + m) * D_EMB + kk;

    v8f accS = {};
    v8f accT = {};

#pragma unroll 4
    for (int k0 = 0; k0 < D_EMB; k0 += 4) {
        v2f a0 = *(const v2f*)(aS + k0);
        v2f b0 = *(const v2f*)(bS + k0);
        accS = __builtin_amdgcn_wmma_f32_16x16x4_f32(
            /*neg_a=*/false, a0, /*neg_b=*/false, b0,
            /*c_mod=*/(short)0, accS, /*reuse_a=*/false, /*reuse_b=*/false);

        v2f a1 = *(const v2f*)(aT + k0);
        v2f b1 = *(const v2f*)(bT + k0);
        accT = __builtin_amdgcn_wmma_f32_16x16x4_f32(
            false, a1, false, b1, (short)0, accT, false, false);
    }

    // C/D layout: VGPR r holds M = r + 8*(lane>=16), N = lane&15.
    const int j      = bj * 16 + (lane & 15);
    const int ibase  = bi * 16 + ((lane >> 4) << 3);
    const float insj = in_s[j], intj = in_t[j];
    const float sqsj = sq_s[j], sqtj = sq_t[j];

    float part = 0.0f;
#pragma unroll
    for (int r = 0; r < 8; ++r) {
        const int i = ibase + r;
        const float gs = accS[r];
        const float gt = accT[r];
        // cosine-similarity term
        const float cs = gs * in_s[i] * insj;
        const float ct = gt * in_t[i] * intj;
        // squared-distance term: sq_i + sq_j - 2*gram
        const float ds = sq_s[i] + sqsj - 2.0f * gs;
        const float dt = sq_t[i] + sqtj - 2.0f * gt;
        const float d0 = cs - ct;
        const float d1 = ds - dt;
        part = fmaf(d0, d0, part);
        part = fmaf(d1, d1, part);
    }

#pragma unroll
    for (int off = 16; off > 0; off >>= 1)
        part += __shfl_xor(part, off, 32);

    if (lane == 0) {
        const float w = (bi == bj) ? 1.0f : 2.0f;   // transpose tile by symmetry
        atomicAdd(acc, w * part);
    }
}

// ---------------------------------------------------------------------------
// Kernel 3: final scaling. DISTANCE_W = ANGLE_W = 1, both terms are means
// over N*N elements.
// ---------------------------------------------------------------------------
__global__ void rkd_finalize(const float* __restrict__ acc,
                             float* __restrict__ out) {
    out[0] = acc[0] * (1.0f / ((float)N_EMB * (float)N_EMB));
}

extern "C" void kernel_launch(void* const* d_in, const int* in_sizes, int n_in,
                              void* d_out, int out_size, void* d_ws, size_t ws_size,
                              hipStream_t stream) {
    const float* s = (const float*)d_in[0];   // student_emb [1024,512] f32
    const float* t = (const float*)d_in[1];   // teacher_emb [1024,512] f32
    float* out = (float*)d_out;               // scalar f32 loss

    // Workspace layout (floats): sq_s | sq_t | inv_s | inv_t | accum
    float* ws    = (float*)d_ws;
    float* sq_s  = ws;
    float* sq_t  = ws + N_EMB;
    float* inv_s = ws + 2 * N_EMB;
    float* inv_t = ws + 3 * N_EMB;
    float* accum = ws + 4 * N_EMB;

    // 2048 rows total, 8 waves (256 threads) per block -> 256 blocks.
    rkd_rowstats<<<256, 256, 0, stream>>>(s, t, sq_s, sq_t, inv_s, inv_t, accum);

    // 64x64 tile grid, one wave32 per tile, upper triangle does the work.
    dim3 grid(N_EMB / 16, N_EMB / 16);
    rkd_gram_tile_loss<<<grid, 32, 0, stream>>>(s, t, sq_s, sq_t, inv_s, inv_t, accum);

    rkd_finalize<<<1, 1, 0, stream>>>(accum, out);
}